// PoVot_77773267796751
// MI455X (gfx1250) — compile-verified
//
#include <hip/hip_runtime.h>
#include <hip/hip_bf16.h>

#define NSEG 512
#define NCH  21

typedef __attribute__((ext_vector_type(2))) float v2f;
typedef __attribute__((ext_vector_type(8))) float v8f;

// ---------------------------------------------------------------------------
// Zero the 512-entry histogram (ws is poisoned, must re-zero every launch).
// ---------------------------------------------------------------------------
__global__ void zero_hist_kernel(int* __restrict__ hist) {
    int g = blockIdx.x * blockDim.x + threadIdx.x;
    if (g < NSEG) hist[g] = 0;
}

// ---------------------------------------------------------------------------
// Build WMMA operand arrays and the segment histogram.
//   p4[j]  = (-2px, -2py, -2pz, |p|^2)   (B matrix source, K=4 trick)
//   dp4[i] = ( dx ,  dy ,  dz , 1.0  )   (A matrix source)
// A.B = |p|^2 - 2 d.p  which has the same argmin over n as the true distance.
// ---------------------------------------------------------------------------
__global__ void prep_kernel(const float* __restrict__ points,
                            const float* __restrict__ d_points,
                            const int*   __restrict__ sptids,
                            float* __restrict__ p4, float* __restrict__ dp4,
                            int* __restrict__ hist, int n, int m) {
    int g = blockIdx.x * blockDim.x + threadIdx.x;
    if (g < n) {
        float x = points[3 * g + 0];
        float y = points[3 * g + 1];
        float z = points[3 * g + 2];
        ((float4*)p4)[g] = make_float4(-2.0f * x, -2.0f * y, -2.0f * z,
                                       x * x + y * y + z * z);
    }
    if (g < m) {
        float x = d_points[3 * g + 0];
        float y = d_points[3 * g + 1];
        float z = d_points[3 * g + 2];
        ((float4*)dp4)[g] = make_float4(x, y, z, 1.0f);
        atomicAdd(&hist[sptids[g]], 1);  // integer atomic: order-independent
    }
}

// ---------------------------------------------------------------------------
// WMMA nearest-neighbor argmin. One wave owns 16 rows of d_points and sweeps
// all n points in 16-wide column tiles: one V_WMMA_F32_16X16X4_F32 per tile
// produces 256 distance surrogates. Lane layout (ISA 7.12.2, 32-bit 16x16
// C/D): lane L, VGPR v holds (M = v + 8*(L>=16), N = L%16).
// A 16x4 f32: lane L holds row M=L%16, components K = 2*(L/16), 2*(L/16)+1.
// B  4x16 f32: lane L holds col N=L%16, rows  K = 2*(L/16), 2*(L/16)+1.
// EXEC is all-ones throughout the WMMA loop (no divergence before it).
// Only ~1024 waves exist for this sweep (possibly ~1/SIMD on MI455X), so the
// tile loop is unrolled x4 to keep 4 clause'd loads + 4 WMMAs in flight
// within a single wave for latency hiding.
// ---------------------------------------------------------------------------
__global__ void __launch_bounds__(256)
nn_kernel(const float* __restrict__ p4, const float* __restrict__ dp4,
          int* __restrict__ nn_idx, int n) {
    const int lane = threadIdx.x & 31;
    const int warp = threadIdx.x >> 5;
    const int r    = lane & 15;   // N (and M) residue within tile
    const int h    = lane >> 4;   // half-wave selector
    const int rowBase = (blockIdx.x * 8 + warp) * 16;

    // A operand: 2 consecutive f32 components of row (rowBase + r)
    const float2 a2 = ((const float2*)(dp4 + 4 * (rowBase + r)))[h];
    v2f a; a[0] = a2.x; a[1] = a2.y;

    float mv[8];
    int   mi[8];
#pragma unroll
    for (int v = 0; v < 8; ++v) { mv[v] = 3.402823466e38f; mi[v] = 0; }

    const int ntiles = n >> 4;
#pragma unroll 4
    for (int t = 0; t < ntiles; ++t) {
        const float2 b2 = ((const float2*)(p4 + 4 * ((t << 4) + r)))[h];
        v2f b; b[0] = b2.x; b[1] = b2.y;
        v8f c = {};
        v8f d = __builtin_amdgcn_wmma_f32_16x16x4_f32(
            /*neg_a=*/false, a, /*neg_b=*/false, b,
            /*c_mod=*/(short)0, c, /*reuse_a=*/false, /*reuse_b=*/false);
        const int col = (t << 4) + r;  // this lane's N column index
#pragma unroll
        for (int v = 0; v < 8; ++v)
            if (d[v] < mv[v]) { mv[v] = d[v]; mi[v] = col; }  // strict '<' keeps first
    }

    // Butterfly reduce across the 16 lanes sharing each row (xor of bits 0..3
    // never crosses the half-wave boundary). Tie-break: lower index wins.
#pragma unroll
    for (int off = 8; off >= 1; off >>= 1) {
#pragma unroll
        for (int v = 0; v < 8; ++v) {
            float ov = __shfl_xor(mv[v], off, 32);
            int   oi = __shfl_xor(mi[v], off, 32);
            if (ov < mv[v] || (ov == mv[v] && oi < mi[v])) { mv[v] = ov; mi[v] = oi; }
        }
    }

    // Row (rowBase + v + 8h) result is in register slot v of lanes r==v.
#pragma unroll
    for (int v = 0; v < 8; ++v)
        if (r == v) nn_idx[rowBase + v + 8 * h] = mi[v];
}

// ---------------------------------------------------------------------------
// Single-block exclusive prefix sum of the 512 segment counts.
// ---------------------------------------------------------------------------
__global__ void scan_kernel(const int* __restrict__ hist, int* __restrict__ offs) {
    __shared__ int s[NSEG];
    int t = threadIdx.x;
    int own = hist[t];
    s[t] = own;
    __syncthreads();
    for (int off = 1; off < NSEG; off <<= 1) {
        int v = (t >= off) ? s[t - off] : 0;
        __syncthreads();
        s[t] += v;
        __syncthreads();
    }
    offs[t] = s[t] - own;  // exclusive
}

// ---------------------------------------------------------------------------
// One wave per segment: stable counting-sort scatter of d_points (ballot +
// popcount preserves index order) fused with a deterministic 21-dim score
// accumulation (fixed lane assignment + fixed xor reduction tree) and the
// per-segment argmax (argmax(sum) == argmax(mean); count is a positive
// per-segment constant, and empty segments yield label 0 like the reference).
// ---------------------------------------------------------------------------
__global__ void __launch_bounds__(256)
segment_kernel(const float* __restrict__ d_points, const int* __restrict__ sptids,
               const int* __restrict__ nn_idx, const float* __restrict__ scores,
               const int* __restrict__ offs, float* __restrict__ out_points,
               int* __restrict__ seg_labels, int m) {
    const int lane = threadIdx.x & 31;
    const int seg  = (blockIdx.x * blockDim.x + threadIdx.x) >> 5;

    int cur = offs[seg];
    float acc[NCH];
#pragma unroll
    for (int c = 0; c < NCH; ++c) acc[c] = 0.0f;

    for (int chunk = 0; chunk < m; chunk += 32) {
        const int i = chunk + lane;                 // m % 32 == 0
        const bool match = (sptids[i] == seg);
        const unsigned long long mb = __ballot(match);
        if (match) {
            const int pos = cur + __popcll(mb & ((1ull << lane) - 1ull));
            out_points[3 * pos + 0] = d_points[3 * i + 0];
            out_points[3 * pos + 1] = d_points[3 * i + 1];
            out_points[3 * pos + 2] = d_points[3 * i + 2];
            const float* sc = scores + (size_t)nn_idx[i] * NCH;
#pragma unroll
            for (int c = 0; c < NCH; ++c) acc[c] += sc[c];
        }
        cur += (int)__popcll(mb);
    }

    // Deterministic cross-lane sum (fixed tree).
#pragma unroll
    for (int off = 16; off >= 1; off >>= 1)
#pragma unroll
        for (int c = 0; c < NCH; ++c) acc[c] += __shfl_xor(acc[c], off, 32);

    if (lane == 0) {
        int best = 0;
        float bv = acc[0];
#pragma unroll
        for (int c = 1; c < NCH; ++c)
            if (acc[c] > bv) { bv = acc[c]; best = c; }  // strict '>' keeps first
        seg_labels[seg] = best;
    }
}

// ---------------------------------------------------------------------------
// labels[i] = seg_labels[sptids[i]]  (int32, stored after the 3m point floats)
// ---------------------------------------------------------------------------
__global__ void labels_kernel(const int* __restrict__ sptids,
                              const int* __restrict__ seg_labels,
                              int* __restrict__ out_labels, int m) {
    int i = blockIdx.x * blockDim.x + threadIdx.x;
    if (i < m) out_labels[i] = seg_labels[sptids[i]];
}

extern "C" void kernel_launch(void* const* d_in, const int* in_sizes, int n_in,
                              void* d_out, int out_size, void* d_ws, size_t ws_size,
                              hipStream_t stream) {
    const float* points   = (const float*)d_in[0];
    const float* scores   = (const float*)d_in[1];
    const float* d_points = (const float*)d_in[2];
    const int*   sptids   = (const int*)d_in[3];
    const int n = in_sizes[0] / 3;   // 32768
    const int m = in_sizes[3];       // 16384

    // Workspace carve-up (~1 MB total).
    char* ws = (char*)d_ws;
    float* p4      = (float*)ws;  ws += (size_t)n * 4 * sizeof(float);
    float* dp4     = (float*)ws;  ws += (size_t)m * 4 * sizeof(float);
    int* nn_idx    = (int*)ws;    ws += (size_t)m * sizeof(int);
    int* hist      = (int*)ws;    ws += NSEG * sizeof(int);
    int* offs      = (int*)ws;    ws += NSEG * sizeof(int);
    int* seg_labels= (int*)ws;

    float* out_points = (float*)d_out;
    int*   out_labels = (int*)d_out + (size_t)3 * m;

    zero_hist_kernel<<<(NSEG + 255) / 256, 256, 0, stream>>>(hist);
    prep_kernel<<<(n + 255) / 256, 256, 0, stream>>>(points, d_points, sptids,
                                                     p4, dp4, hist, n, m);
    // 8 waves/block, 16 rows/wave -> 128 rows/block.
    nn_kernel<<<m / 128, 256, 0, stream>>>(p4, dp4, nn_idx, n);
    scan_kernel<<<1, NSEG, 0, stream>>>(hist, offs);
    // One wave per segment: 512 waves = 64 blocks of 256 threads.
    segment_kernel<<<(NSEG * 32) / 256, 256, 0, stream>>>(
        d_points, sptids, nn_idx, scores, offs, out_points, seg_labels, m);
    labels_kernel<<<(m + 255) / 256, 256, 0, stream>>>(sptids, seg_labels,
                                                       out_labels, m);
}